// depthwise_clipseg_conv_1580547974095
// MI455X (gfx1250) — compile-verified
//
#include <hip/hip_runtime.h>
#include <stdint.h>

// D = A x B + C, f32 16x16x4 WMMA operand types
typedef float v2f __attribute__((ext_vector_type(2)));
typedef float v8f __attribute__((ext_vector_type(8)));

#define HH    128
#define WW    128
#define PITCH 140   // padded LDS row pitch in floats: 4 left pad, 128 interior, 8 right pad
#define PROWS 66    // 1 halo row + 64 interior rows + 1 halo row
#define COL0  4     // padded column index of interior column 0 (keeps b128 LDS writes 16B aligned)

// CDNA5 async global->LDS copy (ASYNCcnt-tracked). Inline asm per CDNA5_HIP.md
// (portable across ROCm7.2 / amdgpu-toolchain; bypasses builtin arity differences).
__device__ __forceinline__ void async_row_load_b128(const float* gsrc, float* ldst) {
  unsigned lofs          = (unsigned)(uintptr_t)ldst;              // low 32 bits of generic ptr == LDS byte offset
  unsigned long long ga  = (unsigned long long)(uintptr_t)gsrc;
  asm volatile("global_load_async_to_lds_b128 %0, %1, off"
               :: "v"(lofs), "v"(ga) : "memory");
}

__device__ __forceinline__ void wait_asynccnt0() {
  asm volatile("s_wait_asynccnt 0" ::: "memory");
}

// One workgroup = one 64-row half of one 128x128 plane.
// Conv expressed as: out = sum_{dy=0..2} A_dy @ T_dy  (banded Toeplitz weights),
// evaluated with v_wmma_f32_16x16x4_f32 on 16x16 output tiles.
__global__ __launch_bounds__(256) void dwconv3x3_wmma(
    const float* __restrict__ x, const float* __restrict__ w9,
    const float* __restrict__ bias, float* __restrict__ out)
{
  __shared__ float smem[PROWS * PITCH];   // 36,960 B

  const int tid  = threadIdx.x;
  const int lane = tid & 31;
  const int wv   = tid >> 5;       // wave 0..7
  const int half = lane >> 4;      // 0: lanes 0-15, 1: lanes 16-31
  const int n    = lane & 15;      // N index (and M index for A rows)

  const int plane = blockIdx.x >> 1;
  const int r0    = (blockIdx.x & 1) << 6;   // first interior row of this half: 0 or 64

  const float* xp = x   + (size_t)plane * (HH * WW);
  float*       op = out + (size_t)plane * (HH * WW);

  // ---- zero halo ring + pad columns (so Toeplitz edge chunks read exact zeros) ----
  if (tid < PITCH) {                               // full top/bottom halo rows
    smem[tid] = 0.0f;
    smem[(PROWS - 1) * PITCH + tid] = 0.0f;
  }
  if (tid < 128) {                                 // pad cols 0..3 and 132..135, rows 1..64
    const int r = 1 + (tid & 63);
    const int c = (tid & 64) ? 132 : 0;
    float4 z = {0.0f, 0.0f, 0.0f, 0.0f};
    *(float4*)(&smem[r * PITCH + c]) = z;          // 16B-aligned ds_store_b128
  }
  __syncthreads();

  // ---- async-stage global rows r0-1 .. r0+64 (clamped) into LDS ----
  // One b128 per lane => one async instruction moves one full 512B row per wave.
  for (int pr = wv; pr < PROWS; pr += 8) {
    const int g = r0 - 1 + pr;
    if (0 <= g && g < HH) {
      async_row_load_b128(xp + g * WW + lane * 4,
                          &smem[pr * PITCH + COL0 + lane * 4]);
    }
  }
  wait_asynccnt0();
  __syncthreads();

  // ---- weights (uniform -> scalar loads) ----
  float w3[9];
#pragma unroll
  for (int i = 0; i < 9; ++i) w3[i] = w9[i];
  const float bv = bias[0];

  // ---- precompute banded-Toeplitz B tiles: B_dy,q[k][n] = w[dy][jglob-n+1], jglob = 4q-2+k
  // Assumed B VGPR layout (mirrors documented A/C patterns): VGPR j, lane L -> k = j + 2*(L>=16), n = L&15.
  v2f bt[3][5];
#pragma unroll
  for (int dy = 0; dy < 3; ++dy) {
#pragma unroll
    for (int q = 0; q < 5; ++q) {
      v2f bb;
#pragma unroll
      for (int j = 0; j < 2; ++j) {
        const int k = j + 2 * half;
        const int d = (4 * q - 2 + k) - n;       // == dx - 1
        float v = 0.0f;
        if (d >= -1 && d <= 1) v = w3[dy * 3 + d + 1];
        bb[j] = v;
      }
      bt[dy][q] = bb;
    }
  }

  // ---- 32 tiles of 16x16 per WG: wave wv owns tile row (wv>>1), 4 tile cols ----
  const int y0  = (wv >> 1) * 16;                  // local row of tile (0..48)
  const int tx0 = (wv & 1) * 4;

#pragma unroll 1
  for (int t = 0; t < 4; ++t) {
    const int x0 = (tx0 + t) * 16;

    v8f acc;                                       // C/D: VGPR r, lane L -> (m = r+8*half, n)
#pragma unroll
    for (int r = 0; r < 8; ++r) acc[r] = bv;       // bias folded into accumulator init

#pragma unroll
    for (int q = 0; q < 5; ++q) {
      // A chunk: lanes hold in[padded_row = y0+m+dy][padded_col = x0 + (4q-2+k) + COL0]
      // v0 -> k = 2*half, v1 -> k = 2*half+1  (documented 16x4 f32 A layout) => one b64 per lane.
      const int cb = x0 + 4 * q + 2 + 2 * half;    // even -> 8B-aligned ds_load_b64
#pragma unroll
      for (int dy = 0; dy < 3; ++dy) {
        const v2f a = *(const v2f*)(&smem[(y0 + n + dy) * PITCH + cb]);
        acc = __builtin_amdgcn_wmma_f32_16x16x4_f32(
            /*neg_a=*/false, a, /*neg_b=*/false, bt[dy][q],
            /*c_mod=*/(short)0, acc, /*reuse_a=*/false, /*reuse_b=*/false);
      }
    }

    // ---- store 16x16 tile (nontemporal: output never re-read, ~size of L2 total) ----
#pragma unroll
    for (int r = 0; r < 8; ++r) {
      const int gy = r0 + y0 + r + 8 * half;
      __builtin_nontemporal_store(acc[r], &op[gy * WW + x0 + n]);
    }
  }
}

extern "C" void kernel_launch(void* const* d_in, const int* in_sizes, int n_in,
                              void* d_out, int out_size, void* d_ws, size_t ws_size,
                              hipStream_t stream) {
  (void)n_in; (void)d_ws; (void)ws_size; (void)out_size;
  const float* x = (const float*)d_in[0];
  const float* w = (const float*)d_in[1];
  const float* b = (const float*)d_in[2];
  float* out = (float*)d_out;
  const int planes = in_sizes[0] / (HH * WW);   // B*C = 3072
  dim3 grid(planes * 2), block(256);
  hipLaunchKernelGGL(dwconv3x3_wmma, grid, block, 0, stream, x, w, b, out);
}